// Attention_Encoder_10797547782158
// MI455X (gfx1250) — compile-verified
//
#include <hip/hip_runtime.h>
#include <math.h>

// Problem sizes (fixed by reference)
#define B_   512
#define T_   256
#define N_   128
#define M_   256
#define M4_  1024   // 4*M
#define K2M_ 512    // 2*M

typedef __attribute__((ext_vector_type(2))) float v2f;
typedef __attribute__((ext_vector_type(8))) float v8f;

#if __has_builtin(__builtin_amdgcn_tanhf)
__device__ __forceinline__ float fast_tanh(float x) { return __builtin_amdgcn_tanhf(x); }
#else
__device__ __forceinline__ float fast_tanh(float x) { return tanhf(x); }
#endif

__device__ __forceinline__ float fast_sigmoid(float x) {
  return 0.5f * fast_tanh(0.5f * x) + 0.5f;
}

// D = A(16x4) * B(4x16) + C using the CDNA5 f32 WMMA path.
__device__ __forceinline__ v8f wmma_f32_4(v2f a, v2f b, v8f c) {
  return __builtin_amdgcn_wmma_f32_16x16x4_f32(false, a, false, b, (short)0, c,
                                               false, false);
}

// ---- gfx1250 async global->LDS copy (ASYNCcnt-tracked), guarded -------------
// Builtin signature (per hipcc diagnostic): arg0 = int __vector(2) addrspace(1)*,
// arg1 = same vector type in LDS (addrspace(3)), arg2/arg3 = imm offset / cpol.
#if defined(__AMDGCN__) && __has_builtin(__builtin_amdgcn_global_load_async_to_lds_b64)
#define ASYNC_LDS 1
typedef int v2i_t __attribute__((vector_size(8)));
typedef __attribute__((address_space(1))) v2i_t g_v2i;   // global (AS1)
typedef __attribute__((address_space(3))) v2i_t l_v2i;   // LDS    (AS3)
__device__ __forceinline__ void async_copy8(void* lds_dst, const void* gsrc) {
  __builtin_amdgcn_global_load_async_to_lds_b64(
      (g_v2i*)(unsigned long long)(uintptr_t)gsrc,
      (l_v2i*)(unsigned)(uintptr_t)lds_dst, 0, 0);
}
__device__ __forceinline__ void async_wait0() {
#if __has_builtin(__builtin_amdgcn_s_wait_asynccnt)
  __builtin_amdgcn_s_wait_asynccnt(0);
#else
  asm volatile("s_wait_asynccnt 0" ::: "memory");
#endif
}
#else
#define ASYNC_LDS 0
#endif

// -----------------------------------------------------------------------------
// Kernel 1: r2[b,n,u] = sum_t x[b,t,n] * Ue[t,u]
// Each wave owns FOUR adjacent 16-wide u-tiles: one A fragment feeds 4 WMMAs
// per K-step (4x fewer A-side loads). Grid: B*(N/16)*(T/64) waves.
// -----------------------------------------------------------------------------
__global__ void __launch_bounds__(128) r2_kernel(const float* __restrict__ x,
                                                 const float* __restrict__ Ue,
                                                 float* __restrict__ r2) {
  const int lane = threadIdx.x & 31;
  const int wave = threadIdx.x >> 5;
  const int g    = blockIdx.x * 4 + wave;   // 512*8*4 = 16384 groups
  const int b    = g >> 5;
  const int rem  = g & 31;
  const int nt   = rem >> 2;                // n tile 0..7
  const int ug   = rem & 3;                 // u group 0..3 (4 tiles each)

  const int m     = lane & 15;
  const int khalf = lane >> 4;

  const float* xb = x + (size_t)b * T_ * N_;
  const int n  = nt * 16 + m;               // A row
  const int uc = ug * 64 + m;               // B col base (4 tiles: +0,16,32,48)

  v8f acc[4];
#pragma unroll
  for (int i = 0; i < 4; ++i) acc[i] = (v8f){};

  for (int k = 0; k < T_; k += 4) {
    const int ka = k + khalf * 2;
    v2f a;
    a.x = xb[(size_t)(ka + 0) * N_ + n];
    a.y = xb[(size_t)(ka + 1) * N_ + n];
    const float* u0 = Ue + (size_t)(ka + 0) * T_ + uc;
    const float* u1 = Ue + (size_t)(ka + 1) * T_ + uc;
#pragma unroll
    for (int i = 0; i < 4; ++i) {
      v2f bb;
      bb.x = u0[16 * i];
      bb.y = u1[16 * i];
      acc[i] = wmma_f32_4(a, bb, acc[i]);
    }
  }

  // C/D layout: VGPR j -> row j + 8*(lane>>4), col = lane&15
  float* outp = r2 + (size_t)b * N_ * T_ + (size_t)(nt * 16) * T_ + ug * 64;
  const int col   = lane & 15;
  const int rbase = khalf * 8;
#pragma unroll
  for (int i = 0; i < 4; ++i)
#pragma unroll
    for (int j = 0; j < 8; ++j)
      outp[(size_t)(rbase + j) * T_ + 16 * i + col] = acc[i][j];
}

// -----------------------------------------------------------------------------
// Kernel 2: LSTM rollout + r1_t = [h_t|s_t] @ We.
// 32 blocks x 16 batch rows, 8 waves. Multi-tile accumulation: z-GEMM does 4
// tiles per A fragment, r1-GEMM 2. x_t tiles staged via async global->LDS.
// -----------------------------------------------------------------------------
#define HS_STRIDE 521   // 521 % 64 = 9, coprime -> conflict-free column reads
#define Z_STRIDE  1033  // 1033 % 64 = 9
#define XT_STRIDE 130   // 130 % 64 = 2 -> 2m distinct mod 64; 8B-aligned rows

__global__ void __launch_bounds__(256) lstm_rollout_kernel(
    const float* __restrict__ x, const float* __restrict__ s0,
    const float* __restrict__ h0, const float* __restrict__ We,
    const float* __restrict__ Wk, const float* __restrict__ Wr,
    const float* __restrict__ bias, float* __restrict__ R1) {
  __shared__ __align__(16) float hs[16][HS_STRIDE];  // [h | s] per batch row
  __shared__ __align__(16) float zsh[16][Z_STRIDE];  // gate pre-activations
  __shared__ __align__(16) float xt[16][XT_STRIDE];  // x_t tile [16][128]

  const int tid   = threadIdx.x;
  const int lane  = tid & 31;
  const int wave  = tid >> 5;
  const int b0    = blockIdx.x * 16;
  const int m     = lane & 15;
  const int khalf = lane >> 4;
  const int col   = lane & 15;
  const int rbase = khalf * 8;

  // Init carry: hs = [h0 | s0]
  for (int i = tid; i < 16 * K2M_; i += 256) {
    const int r = i >> 9, cidx = i & 511;
    hs[r][cidx] = (cidx < M_) ? h0[(size_t)(b0 + r) * M_ + cidx]
                              : s0[(size_t)(b0 + r) * M_ + (cidx - M_)];
  }
  __syncthreads();

  for (int t = 0; t < T_; ++t) {
    // Stage x_t tile (16 rows x 128 floats) directly into LDS.
#if ASYNC_LDS
    for (int i = tid; i < 16 * (N_ / 2); i += 256) {
      const int r = i >> 6, q = (i & 63) * 2;
      async_copy8(&xt[r][q], &x[((size_t)(b0 + r) * T_ + t) * N_ + q]);
    }
    async_wait0();
#else
    for (int i = tid; i < 16 * (N_ / 2); i += 256) {
      const int r = i >> 6, q = (i & 63) * 2;
      const float2 v = *(const float2*)&x[((size_t)(b0 + r) * T_ + t) * N_ + q];
      xt[r][q] = v.x;
      xt[r][q + 1] = v.y;
    }
#endif
    // Hint next step's x_t tile toward the caches while we compute.
    if (t + 1 < T_ && tid < 16)
      __builtin_prefetch(&x[((size_t)(b0 + tid) * T_ + (t + 1)) * N_], 0, 0);
    __syncthreads();

    // r1_t = [h|s] @ We (ENTERING carry); each wave: 2 u-tiles, shared A frag.
    {
      const int uc = wave * 32 + m;
      v8f c0 = {}, c1 = {};
      for (int k = 0; k < K2M_; k += 4) {
        const int ka = k + khalf * 2;
        v2f a;
        a.x = hs[m][ka];
        a.y = hs[m][ka + 1];
        const float* w0 = We + (size_t)(ka + 0) * T_ + uc;
        const float* w1 = We + (size_t)(ka + 1) * T_ + uc;
        v2f bb0, bb1;
        bb0.x = w0[0];  bb0.y = w1[0];
        bb1.x = w0[16]; bb1.y = w1[16];
        c0 = wmma_f32_4(a, bb0, c0);
        c1 = wmma_f32_4(a, bb1, c1);
      }
      float* outp = R1 + ((size_t)t * B_ + b0) * T_ + wave * 32;
#pragma unroll
      for (int j = 0; j < 8; ++j) {
        outp[(size_t)(rbase + j) * T_ + col]      = c0[j];
        outp[(size_t)(rbase + j) * T_ + 16 + col] = c1[j];
      }
    }

    // z = x_t @ Wk + h @ Wr + bias : [16][1024]. 16 groups of 4 tiles.
    for (int ng = wave; ng < 16; ng += 8) {
      const int nc = ng * 64 + m;
      v8f acc[4];
#pragma unroll
      for (int i = 0; i < 4; ++i) {
        const float bv = bias[nc + 16 * i];
#pragma unroll
        for (int j = 0; j < 8; ++j) acc[i][j] = bv;
      }
      for (int k = 0; k < N_; k += 4) {   // x_t @ Wk
        const int ka = k + khalf * 2;
        v2f a;
        a.x = xt[m][ka];
        a.y = xt[m][ka + 1];
        const float* w0 = Wk + (size_t)(ka + 0) * M4_ + nc;
        const float* w1 = Wk + (size_t)(ka + 1) * M4_ + nc;
#pragma unroll
        for (int i = 0; i < 4; ++i) {
          v2f bb;
          bb.x = w0[16 * i];
          bb.y = w1[16 * i];
          acc[i] = wmma_f32_4(a, bb, acc[i]);
        }
      }
      for (int k = 0; k < M_; k += 4) {   // h @ Wr
        const int ka = k + khalf * 2;
        v2f a;
        a.x = hs[m][ka];
        a.y = hs[m][ka + 1];
        const float* w0 = Wr + (size_t)(ka + 0) * M4_ + nc;
        const float* w1 = Wr + (size_t)(ka + 1) * M4_ + nc;
#pragma unroll
        for (int i = 0; i < 4; ++i) {
          v2f bb;
          bb.x = w0[16 * i];
          bb.y = w1[16 * i];
          acc[i] = wmma_f32_4(a, bb, acc[i]);
        }
      }
#pragma unroll
      for (int i = 0; i < 4; ++i)
#pragma unroll
        for (int j = 0; j < 8; ++j)
          zsh[rbase + j][ng * 64 + 16 * i + col] = acc[i][j];
    }
    __syncthreads();

    // Gate math (keras order i,f,g,o); each thread owns its (r,j) cells.
    for (int i = tid; i < 16 * M_; i += 256) {
      const int r = i >> 8, j = i & 255;
      const float ig = fast_sigmoid(zsh[r][j]);
      const float fg = fast_sigmoid(zsh[r][M_ + j]);
      const float g  = fast_tanh(zsh[r][2 * M_ + j]);
      const float og = fast_sigmoid(zsh[r][3 * M_ + j]);
      const float sv = fg * hs[r][M_ + j] + ig * g;
      hs[r][M_ + j] = sv;                 // s_{t+1}
      hs[r][j]      = og * fast_tanh(sv); // h_{t+1}
    }
    __syncthreads();
  }
}

// -----------------------------------------------------------------------------
// Kernel 3: e[b,n] = sum_u tanh(r1_t[b,u] + r2[b,n,u]) * ve[u]; softmax over n;
// out = alpha * x. One block per batch row; r2[b] (129KB) async-staged into LDS
// once and reused across all 256 steps (saves ~17 GB of HBM re-reads).
// tanh-bound -> native v_tanh_f32.
// -----------------------------------------------------------------------------
#define R2S 258  // 258 % 64 = 2 -> 2n distinct mod 64 for 32 lanes; 8B-aligned

__global__ void __launch_bounds__(256) attn_kernel(
    const float* __restrict__ x, const float* __restrict__ ve,
    const float* __restrict__ r2, const float* __restrict__ R1,
    float* __restrict__ out) {
  __shared__ __align__(16) float r2s[N_][R2S];
  __shared__ float r1s[T_];
  __shared__ float ves[T_];
  __shared__ float part[256];
  __shared__ float esh[N_];
  __shared__ float red[64];

  const int tid = threadIdx.x;
  const int b   = blockIdx.x;

  const float* r2b = r2 + (size_t)b * N_ * T_;
#if ASYNC_LDS
  for (int i = tid; i < (N_ * T_) / 2; i += 256) {
    const int n = i >> 7, q = (i & 127) * 2;
    async_copy8(&r2s[n][q], r2b + (size_t)n * T_ + q);
  }
  async_wait0();
#else
  for (int i = tid; i < (N_ * T_) / 2; i += 256) {
    const int n = i >> 7, q = (i & 127) * 2;
    const float2 v = *(const float2*)(r2b + (size_t)n * T_ + q);
    r2s[n][q] = v.x;
    r2s[n][q + 1] = v.y;
  }
#endif
  ves[tid] = ve[tid];
  __syncthreads();

  const int n    = tid & 127;
  const int half = tid >> 7;
  const int u0   = half * 128;

  for (int t = 0; t < T_; ++t) {
    r1s[tid] = R1[((size_t)t * B_ + b) * T_ + tid];
    if (t + 1 < T_ && (tid & 31) == 0)
      __builtin_prefetch(&R1[((size_t)(t + 1) * B_ + b) * T_ + tid], 0, 0);
    __syncthreads();

    float acc = 0.f;
#pragma unroll 4
    for (int u = 0; u < 128; ++u) {
      const int uu = u0 + u;
      acc += fast_tanh(r1s[uu] + r2s[n][uu]) * ves[uu];
    }
    part[tid] = acc;
    __syncthreads();
    if (half == 0) esh[n] = part[n] + part[n + 128];
    __syncthreads();

    // softmax over n=128: max
    if (tid < 64) red[tid] = fmaxf(esh[tid], esh[tid + 64]);
    __syncthreads();
    for (int sft = 32; sft >= 1; sft >>= 1) {
      if (tid < sft) red[tid] = fmaxf(red[tid], red[tid + sft]);
      __syncthreads();
    }
    const float emax = red[0];
    __syncthreads();

    if (half == 0) esh[n] = __expf(esh[n] - emax);
    __syncthreads();
    if (tid < 64) red[tid] = esh[tid] + esh[tid + 64];
    __syncthreads();
    for (int sft = 32; sft >= 1; sft >>= 1) {
      if (tid < sft) red[tid] += red[tid + sft];
      __syncthreads();
    }
    const float inv = 1.0f / red[0];

    if (half == 0) {
      const size_t idx = ((size_t)b * T_ + t) * N_ + n;
      out[idx] = esh[n] * inv * x[idx];
    }
    __syncthreads();
  }
}

// -----------------------------------------------------------------------------
extern "C" void kernel_launch(void* const* d_in, const int* in_sizes, int n_in,
                              void* d_out, int out_size, void* d_ws,
                              size_t ws_size, hipStream_t stream) {
  (void)in_sizes; (void)n_in; (void)out_size; (void)ws_size;
  const float* x    = (const float*)d_in[0];
  const float* s0   = (const float*)d_in[1];
  const float* h0   = (const float*)d_in[2];
  const float* We   = (const float*)d_in[3];
  const float* Ue   = (const float*)d_in[4];
  const float* ve   = (const float*)d_in[5];
  const float* Wk   = (const float*)d_in[6];
  const float* Wr   = (const float*)d_in[7];
  const float* bias = (const float*)d_in[8];
  float* out = (float*)d_out;

  float* r2 = (float*)d_ws;                       // B*N*T floats (64 MB)
  float* R1 = r2 + (size_t)B_ * N_ * T_;          // T*B*T floats (128 MB)

  // 1) r2 precompute (independent of the scan)
  r2_kernel<<<(B_ * (N_ / 16) * (T_ / 64)) / 4, 128, 0, stream>>>(x, Ue, r2);
  // 2) LSTM rollout + r1_t for all t (parallel over batch, sequential in t)
  lstm_rollout_kernel<<<B_ / 16, 256, 0, stream>>>(x, s0, h0, We, Wk, Wr, bias, R1);
  // 3) scores + softmax + output (tanh-bound; r2[b] LDS-resident)
  attn_kernel<<<B_, 256, 0, stream>>>(x, ve, r2, R1, out);
}